// SpectralAttention_65987877536047
// MI455X (gfx1250) — compile-verified
//
#include <hip/hip_runtime.h>

// SpectralAttention fused kernel for MI455X (gfx1250, wave32).
// scores[b,s,a] = PReLU( <input[b,a,:], w1> + <scattered[b,s,a,:], w2> )
// out = softmax over s (S=3).
// Memory-bound: streams 1.07 GB once; dot products run on the matrix pipe
// via V_WMMA_F32_16X16X4_F32 (full fp32 precision).

typedef __attribute__((ext_vector_type(2))) float v2f;
typedef __attribute__((ext_vector_type(4))) float v4f;
typedef __attribute__((ext_vector_type(8))) float v8f;

#define NB 1024
#define NS 3
#define NA 256
#define ND 256

__global__ __launch_bounds__(256) void spectral_attn_65987877536047(
    const float* __restrict__ input,      // [B, A, D]
    const float* __restrict__ scattered,  // [B, S, A, D]
    const float* __restrict__ wattn,      // [2*D] = {w1, w2}
    const float* __restrict__ prelu,      // [1]
    float* __restrict__ out)              // [B, S, A]
{
    __shared__ __align__(16) float lds_w[2 * ND];            // 2 KB weights
    __shared__ __align__(16) float lds_scores[(NS + 1) * NA]; // 4 KB: [s_in | s_sc0 | s_sc1 | s_sc2]

    const int tid  = threadIdx.x;       // 0..255 (8 wave32 waves)
    const int b    = blockIdx.x;        // one block per batch element
    const int lane = tid & 31;
    const int wave = tid >> 5;          // 0..7

    // Stage the 512 attention weights into LDS once per block.
    for (int i = tid; i < 2 * ND; i += 256) lds_w[i] = wattn[i];
    __syncthreads();

    // WMMA A-tile addressing (32-bit A, 16x4):
    //   lanes 0-15  hold rows M=0..15, K pair {0,1}
    //   lanes 16-31 hold rows M=0..15, K pair {2,3}
    const int m    = lane & 15;         // row within 16-row tile
    const int koff = (lane >> 4) << 1;  // 0 or 2: K sub-column offset (floats)

    // 1024 rows per b (256 input rows + 768 scattered rows) = 64 tiles of 16.
    // 8 waves x 8 tiles each. Tiles 0..15 -> input/w1, tiles 16..63 -> scattered/w2.
    for (int i = 0; i < 8; ++i) {
        const int tile = wave * 8 + i;
        const int r0   = tile * 16;     // first global row of the tile

        const float* rowbase;
        int wbase;
        if (r0 < NA) {
            rowbase = input + ((size_t)b * NA + (size_t)(r0 + m)) * ND;
            wbase   = 0;                // w1
        } else {
            rowbase = scattered + ((size_t)b * (NS * NA) + (size_t)(r0 - NA + m)) * ND;
            wbase   = ND;               // w2
        }
        const float* rp = rowbase + koff;            // per-lane global row ptr
        const float* wp = lds_w + wbase + koff;      // broadcast weight chunk

        v8f c = {};
        // K loop: 64 steps of K=4. Each lane: one global_load_b64 (16B/row/step,
        // contiguous -> every 64B line consumed over 4 steps) + one ds_load_b64
        // (lanes 0-15 and 16-31 each broadcast-read the same 8B -> conflict-free).
#pragma unroll 8
        for (int k = 0; k < ND; k += 4) {
            v2f av = *(const v2f*)(rp + k);
            v2f bv = *(const v2f*)(wp + k);
            // D = A(16x4 f32) x B(4x16 f32) + C ; B columns identical, so
            // column 0 of D carries the 16 row-dots for this K chunk.
            c = __builtin_amdgcn_wmma_f32_16x16x4_f32(
                    false, av, false, bv, (short)0, c, false, false);
        }

        // C/D layout: lane 0 (N=0) holds rows 0..7 in c[0..7];
        //             lane 16 (N=0) holds rows 8..15 in c[0..7].
        if (m == 0) {
            const int base = r0 + (lane >> 4) * 8;
            *(v4f*)(&lds_scores[base])     = (v4f){c[0], c[1], c[2], c[3]};
            *(v4f*)(&lds_scores[base + 4]) = (v4f){c[4], c[5], c[6], c[7]};
        }
    }
    __syncthreads();

    // Phase 2: PReLU + softmax over S=3, one thread per 'a'. Coalesced stores.
    const float alpha = prelu[0];
    const int a = tid;
    const float si = lds_scores[a];
    float x0 = si + lds_scores[1 * NA + a];
    float x1 = si + lds_scores[2 * NA + a];
    float x2 = si + lds_scores[3 * NA + a];
    x0 = (x0 >= 0.0f) ? x0 : alpha * x0;
    x1 = (x1 >= 0.0f) ? x1 : alpha * x1;
    x2 = (x2 >= 0.0f) ? x2 : alpha * x2;
    const float mx = fmaxf(x0, fmaxf(x1, x2));
    const float e0 = __expf(x0 - mx);
    const float e1 = __expf(x1 - mx);
    const float e2 = __expf(x2 - mx);
    const float inv = 1.0f / (e0 + e1 + e2);
    float* ob = out + (size_t)b * (NS * NA) + a;
    ob[0 * NA] = e0 * inv;
    ob[1 * NA] = e1 * inv;
    ob[2 * NA] = e2 * inv;
}

extern "C" void kernel_launch(void* const* d_in, const int* in_sizes, int n_in,
                              void* d_out, int out_size, void* d_ws, size_t ws_size,
                              hipStream_t stream) {
    const float* input     = (const float*)d_in[0];  // B*A*D fp32
    const float* scattered = (const float*)d_in[1];  // B*S*A*D fp32
    const float* wattn     = (const float*)d_in[2];  // 2*D fp32
    const float* prelu     = (const float*)d_in[3];  // 1 fp32
    float* out = (float*)d_out;                      // B*S*A fp32

    spectral_attn_65987877536047<<<NB, 256, 0, stream>>>(
        input, scattered, wattn, prelu, out);
}